// SAEHDLightLoss_43430709297530
// MI455X (gfx1250) — compile-verified
//
#include <hip/hip_runtime.h>
#include <math.h>

typedef __attribute__((ext_vector_type(2))) float v2f;
typedef __attribute__((ext_vector_type(8))) float v8f;

#define N_IMG 48
#define CCH 3
#define HH 256
#define WW 256
#define KS 11
#define HALO 5
#define TILE_R 16
#define TILE_C 64
#define SROWS 28            // 26 valid halo rows + 2 zero rows (K padded to 28)
#define SPITCH 72           // bank-conflict-free pitch for half-wave row-pair reads
#define BLK 128             // 4 waves of 32
#define ROW_BANDS (HH / TILE_R)                    // 16
#define COL_BANDS (WW / TILE_C)                    // 4
#define TILES_PER_IMG (CCH * ROW_BANDS * COL_BANDS) // 192
#define NBLOCKS (N_IMG * TILES_PER_IMG)            // 9216

// Gaussian tap, sigma=1.5, normalized by ginv; zero outside [0, KS)
__device__ __forceinline__ float gval(int d, float ginv) {
    if (d < 0 || d >= KS) return 0.0f;
    float x = (float)d - 5.0f;
    return expf(-x * x / 4.5f) * ginv;
}

// Point-in-polygon test replicating the reference's crossing rule + floored bbox.
__device__ __forceinline__ float inPoly(int p, float X, float Y,
                                        const float (*sPts)[2],
                                        const float (*sBB)[5]) {
    float xmn = sBB[p][0], xmx = sBB[p][1], ymn = sBB[p][2], ymx = sBB[p][3];
    float valid = sBB[p][4];
    bool bbox = (Y >= ymn) && (Y < ymx) && (X >= xmn) && (X < xmx);
    int b0 = (p == 0) ? 0 : ((p == 1) ? 6 : 12);
    int K = (p == 2) ? 12 : 6;
    int cross = 0;
    for (int s = 0; s < K; ++s) {
        float x1 = sPts[b0 + s][0], y1 = sPts[b0 + s][1];
        int s2 = (s + 1 == K) ? 0 : (s + 1);
        float x2 = sPts[b0 + s2][0], y2 = sPts[b0 + s2][1];
        bool crossing = ((y1 > Y) != (y2 > Y)) &&
                        (X < (x2 - x1) * (Y - y1) / (y2 - y1 + 1e-6f) + x1);
        cross += crossing ? 1 : 0;
    }
    bool inside = (cross & 1) != 0;
    return (valid != 0.0f && inside && bbox) ? 1.0f : 0.0f;
}

__global__ __launch_bounds__(BLK) void ssim_l1_kernel(
    const float* __restrict__ pred, const float* __restrict__ targ,
    const float* __restrict__ lmk, float* __restrict__ partials) {
    __shared__ float sH[5][SROWS][SPITCH];
    __shared__ float sPts[24][2];
    __shared__ float sBB[3][5];
    __shared__ float sRed[3][BLK];

    const int b = blockIdx.x;
    const int tileIdx = b % (ROW_BANDS * COL_BANDS);
    const int c = (b / (ROW_BANDS * COL_BANDS)) % CCH;
    const int n = b / TILES_PER_IMG;
    const int rowbase = (tileIdx / COL_BANDS) * TILE_R;
    const int colbase = (tileIdx % COL_BANDS) * TILE_C;
    const int tid = threadIdx.x;

    // Gaussian taps (exact f32 recompute, matches reference construction)
    float g[KS];
    float gsum = 0.0f;
#pragma unroll
    for (int t = 0; t < KS; ++t) {
        float x = (float)t - 5.0f;
        g[t] = expf(-x * x / 4.5f);
        gsum += g[t];
    }
    const float ginv = 1.0f / gsum;
#pragma unroll
    for (int t = 0; t < KS; ++t) g[t] *= ginv;

    // Stage landmarks 36..59 (eyeL 6, eyeR 6, mouth 12) into LDS
    if (tid < 24) {
        sPts[tid][0] = lmk[n * 136 + 2 * (36 + tid) + 0];
        sPts[tid][1] = lmk[n * 136 + 2 * (36 + tid) + 1];
    }
    __syncthreads();
    if (tid < 3) {
        int b0 = (tid == 0) ? 0 : ((tid == 1) ? 6 : 12);
        int K = (tid == 2) ? 12 : 6;
        float xmn = sPts[b0][0], xmx = xmn, ymn = sPts[b0][1], ymx = ymn;
        for (int s = 1; s < K; ++s) {
            float xv = sPts[b0 + s][0], yv = sPts[b0 + s][1];
            xmn = fminf(xmn, xv); xmx = fmaxf(xmx, xv);
            ymn = fminf(ymn, yv); ymx = fmaxf(ymx, yv);
        }
        xmn = floorf(xmn); xmx = floorf(xmx);
        ymn = floorf(ymn); ymx = floorf(ymx);
        float valid = (xmn >= 0.0f && ymn >= 0.0f &&
                       xmx < (float)WW && ymx < (float)HH) ? 1.0f : 0.0f;
        sBB[tid][0] = xmn; sBB[tid][1] = xmx;
        sBB[tid][2] = ymn; sBB[tid][3] = ymx; sBB[tid][4] = valid;
    }

    const float* xp = pred + (size_t)(n * CCH + c) * HH * WW;
    const float* yp = targ + (size_t)(n * CCH + c) * HH * WW;

    // ---- Phase 1: horizontal 11-tap blur of 5 maps into LDS (26 halo rows) ----
    for (int p = tid; p < SROWS * TILE_C; p += BLK) {
        int k = p >> 6;
        int j = p & 63;
        float hx = 0, hy = 0, hxx = 0, hyy = 0, hxy = 0;
        int r = rowbase + k - HALO;
        if (k < 26 && r >= 0 && r < HH) {
            const float* xr = xp + r * WW;
            const float* yr = yp + r * WW;
            int c0 = colbase + j - HALO;
#pragma unroll
            for (int t = 0; t < KS; ++t) {
                int cc = c0 + t;
                if (cc >= 0 && cc < WW) {
                    float xv = xr[cc], yv = yr[cc], w = g[t];
                    hx += w * xv; hy += w * yv;
                    hxx += w * xv * xv; hyy += w * yv * yv; hxy += w * xv * yv;
                }
            }
        }
        sH[0][k][j] = hx; sH[1][k][j] = hy;
        sH[2][k][j] = hxx; sH[3][k][j] = hyy; sH[4][k][j] = hxy;
    }
    __syncthreads();

    // ---- Phase 2: vertical pass as banded matmul on the matrix pipe ----
    // O[16x16] = Gv[16x28] * Htmp[28x16], 7 chained V_WMMA_F32_16X16X4_F32.
    const int lane = tid & 31;
    const int wv = tid >> 5;          // wave -> 16-col sub-tile
    const int nlo = lane & 15;
    const int hi = (lane >> 4) & 1;
    const int cidx = wv * 16 + nlo;   // local column 0..63

    // A fragments: lane(M=nlo) holds K = 4s+2*hi, 4s+2*hi+1 (ISA 16x4 f32 layout)
    v2f afrag[7];
#pragma unroll
    for (int s = 0; s < 7; ++s) {
        int kr = 4 * s + 2 * hi;
        v2f a;
        a.x = gval(kr - nlo, ginv);       // Gv[m][k] = g[k-m]
        a.y = gval(kr + 1 - nlo, ginv);
        afrag[s] = a;
    }

    v8f acc0 = {}, acc1 = {}, acc2 = {}, acc3 = {}, acc4 = {};
#pragma unroll
    for (int s = 0; s < 7; ++s) {
        int r0 = 4 * s + 2 * hi;
        v2f a = afrag[s];
        v2f b0; b0.x = sH[0][r0][cidx]; b0.y = sH[0][r0 + 1][cidx];
        v2f b1; b1.x = sH[1][r0][cidx]; b1.y = sH[1][r0 + 1][cidx];
        v2f b2; b2.x = sH[2][r0][cidx]; b2.y = sH[2][r0 + 1][cidx];
        v2f b3; b3.x = sH[3][r0][cidx]; b3.y = sH[3][r0 + 1][cidx];
        v2f b4; b4.x = sH[4][r0][cidx]; b4.y = sH[4][r0 + 1][cidx];
        acc0 = __builtin_amdgcn_wmma_f32_16x16x4_f32(false, a, false, b0, (short)0, acc0, false, false);
        acc1 = __builtin_amdgcn_wmma_f32_16x16x4_f32(false, a, false, b1, (short)0, acc1, false, false);
        acc2 = __builtin_amdgcn_wmma_f32_16x16x4_f32(false, a, false, b2, (short)0, acc2, false, false);
        acc3 = __builtin_amdgcn_wmma_f32_16x16x4_f32(false, a, false, b3, (short)0, acc3, false, false);
        acc4 = __builtin_amdgcn_wmma_f32_16x16x4_f32(false, a, false, b4, (short)0, acc4, false, false);
    }

    // ---- SSIM algebra + L1 + landmark-weighted L1 (VALU, overlaps matrix pipe) ----
    const float C1 = 0.0001f, C2 = 0.0009f;
    float dsum = 0, l1sum = 0, lwsum = 0;
    const int col = colbase + cidx;
    const float X = (float)col;
#pragma unroll
    for (int v = 0; v < 8; ++v) {
        int R = rowbase + v + 8 * hi;  // D layout: VGPR v -> M = v + 8*hi
        float mux = acc0[v], muy = acc1[v];
        float exx = acc2[v], eyy = acc3[v], exy = acc4[v];
        float sx = exx - mux * mux;
        float sy = eyy - muy * muy;
        float sxy = exy - mux * muy;
        float num = (2.0f * mux * muy + C1) * (2.0f * sxy + C2);
        float den = (mux * mux + muy * muy + C1) * (sx + sy + C2);
        float ssim = num / (den + 1e-8f);
        dsum += 0.5f * (1.0f - ssim);
        float xv = xp[R * WW + col], yv = yp[R * WW + col];
        float ad = fabsf(xv - yv);
        l1sum += ad;
        float Y = (float)R;
        float w = 1.0f + 3.0f * inPoly(0, X, Y, sPts, sBB)
                       + 3.0f * inPoly(1, X, Y, sPts, sBB)
                       + 2.0f * inPoly(2, X, Y, sPts, sBB);
        lwsum += w * ad;
    }

    // ---- Block reduction -> per-block partials (deterministic) ----
    sRed[0][tid] = dsum; sRed[1][tid] = l1sum; sRed[2][tid] = lwsum;
    __syncthreads();
    for (int st = BLK / 2; st > 0; st >>= 1) {
        if (tid < st) {
            sRed[0][tid] += sRed[0][tid + st];
            sRed[1][tid] += sRed[1][tid + st];
            sRed[2][tid] += sRed[2][tid + st];
        }
        __syncthreads();
    }
    if (tid == 0) {
        partials[b * 3 + 0] = sRed[0][0];
        partials[b * 3 + 1] = sRed[1][0];
        partials[b * 3 + 2] = sRed[2][0];
    }
}

__global__ void finalize_kernel(const float* __restrict__ partials,
                                float* __restrict__ out) {
    __shared__ float sl[64];
    int n = threadIdx.x;
    float li = 0.0f;
    if (n < N_IMG) {
        float d = 0, l1 = 0, lw = 0;
        int base = n * TILES_PER_IMG;
        for (int t = 0; t < TILES_PER_IMG; ++t) {
            d  += partials[(base + t) * 3 + 0];
            l1 += partials[(base + t) * 3 + 1];
            lw += partials[(base + t) * 3 + 2];
        }
        const float inv = 1.0f / (float)(CCH * HH * WW);
        li = 10.0f * d * inv + 10.0f * l1 * inv + 5.0f * lw * inv;
    }
    sl[n] = li;
    __syncthreads();
    for (int st = 32; st > 0; st >>= 1) {
        if (n < st) sl[n] += sl[n + st];
        __syncthreads();
    }
    if (n == 0) out[0] = sl[0] / (float)N_IMG;
}

extern "C" void kernel_launch(void* const* d_in, const int* in_sizes, int n_in,
                              void* d_out, int out_size, void* d_ws, size_t ws_size,
                              hipStream_t stream) {
    const float* pred = (const float*)d_in[0];
    const float* targ = (const float*)d_in[1];
    const float* lmk  = (const float*)d_in[2];
    float* partials = (float*)d_ws;  // NBLOCKS*3 floats (~110 KB)
    ssim_l1_kernel<<<NBLOCKS, BLK, 0, stream>>>(pred, targ, lmk, partials);
    finalize_kernel<<<1, 64, 0, stream>>>(partials, (float*)d_out);
}